// edgeGAT_36867999268938
// MI455X (gfx1250) — compile-verified
//
#include <hip/hip_runtime.h>
#include <math.h>

typedef float v2f __attribute__((ext_vector_type(2)));
typedef float v8f __attribute__((ext_vector_type(8)));

#define NEG_SLOPE 0.2f

// ---- CDNA5 async global->LDS staging (guarded: falls back to ld+st) ---------
#if defined(__AMDGCN__) &&                                                     \
    __has_builtin(__builtin_amdgcn_global_load_async_to_lds_b128) &&           \
    __has_builtin(__builtin_amdgcn_s_wait_asynccnt)
#define USE_ASYNC 1
#else
#define USE_ASYNC 0
#endif

#if USE_ASYNC
typedef int v4i_vs __attribute__((vector_size(16)));
#define AS1 __attribute__((address_space(1)))
#define AS3 __attribute__((address_space(3)))
__device__ __forceinline__ void async_ld16(const float* g, float* l) {
  // global_load_async_to_lds_b128: (AS1 v4i* gaddr, AS3 v4i* ldsaddr, imm off,
  // imm cpol); tracked with ASYNCcnt.
  __builtin_amdgcn_global_load_async_to_lds_b128((AS1 v4i_vs*)g,
                                                 (AS3 v4i_vs*)l, 0, 0);
}
__device__ __forceinline__ void async_wait0() {
  __builtin_amdgcn_s_wait_asynccnt(0);
}
#endif

// ---------------- WMMA helper: D = A(16x4) * B(4x16) + C, fp32 ----------------
__device__ __forceinline__ v8f wmma4(v2f a, v2f b, v8f c) {
  // 8 args: (neg_a, A, neg_b, B, c_mod, C, reuse_a, reuse_b)
  return __builtin_amdgcn_wmma_f32_16x16x4_f32(false, a, false, b, (short)0, c,
                                               false, false);
}

__device__ __forceinline__ float lrelu(float x) {
  return x >= 0.f ? x : NEG_SLOPE * x;
}
// monotone float->uint key for atomicMax-based segment max
__device__ __forceinline__ unsigned fkey(float f) {
  unsigned b = __float_as_uint(f);
  return (b & 0x80000000u) ? ~b : (b | 0x80000000u);
}
__device__ __forceinline__ float funkey(unsigned k) {
  unsigned b = (k & 0x80000000u) ? (k & 0x7fffffffu) : ~k;
  return __uint_as_float(b);
}

// ---------------- generic GEMM: out[N,64] = X[N,K] @ W[K,64] (+bias) ----------
// block = 128 threads (4 waves); block computes 64 rows x 64 cols.
// wave w -> rows [w*16, w*16+16), 4 col tiles each, K-loop of 16x16x4 WMMA.
// A staged in LDS (stride 132 -> conflict-free, rows 16B-aligned);
// W streamed through WGP$ (<=32KB, fully cache-resident).
__global__ __launch_bounds__(128) void gemm_k_n64(
    const float* __restrict__ X, const float* __restrict__ W,
    const float* __restrict__ bias, float* __restrict__ out,
    int nrows, int K) {
  __shared__ float Alds[64 * 132];  // 33792 B
  const int t = threadIdx.x;
  const int kshift = (K == 128) ? 7 : 6;
  const int row0 = blockIdx.x * 64;

#if USE_ASYNC
  {
    const int c4n = K >> 2;  // 16B chunks per row (32 or 16)
    const int csh = kshift - 2;
    for (int i = t; i < (c4n << 6); i += 128) {
      int r = i >> csh, c = (i & (c4n - 1)) << 2;
      int gr = row0 + r;
      float* l = &Alds[r * 132 + c];
      if (gr < nrows) {
        async_ld16(&X[(size_t)gr * K + c], l);
      } else {
        l[0] = 0.f; l[1] = 0.f; l[2] = 0.f; l[3] = 0.f;
      }
    }
    async_wait0();
  }
#else
  {
    const int kmask = K - 1;
    for (int i = t; i < (K << 6); i += 128) {
      int r = i >> kshift, kk = i & kmask;
      int gr = row0 + r;
      Alds[r * 132 + kk] = (gr < nrows) ? X[(size_t)gr * K + kk] : 0.f;
    }
  }
#endif
  __syncthreads();

  const int wave = t >> 5, lane = t & 31;
  const int half = lane >> 4, l16 = lane & 15;
  const int rbase = wave * 16;
  const int arow = rbase + l16;

  v8f zero = {0.f, 0.f, 0.f, 0.f, 0.f, 0.f, 0.f, 0.f};
  v8f acc[4];
#pragma unroll
  for (int i = 0; i < 4; ++i) acc[i] = zero;

  for (int k0 = 0; k0 < K; k0 += 4) {
    const int ka = k0 + half * 2;
    v2f a = {Alds[arow * 132 + ka], Alds[arow * 132 + ka + 1]};
#pragma unroll
    for (int ct = 0; ct < 4; ++ct) {
      v2f b = {W[ka * 64 + ct * 16 + l16], W[(ka + 1) * 64 + ct * 16 + l16]};
      acc[ct] = wmma4(a, b, acc[ct]);
    }
  }

#pragma unroll
  for (int ct = 0; ct < 4; ++ct) {
    const int col = ct * 16 + l16;
    const float badd = bias ? bias[col] : 0.f;
#pragma unroll
    for (int v = 0; v < 8; ++v) {
      int gr = row0 + rbase + half * 8 + v;  // C layout: vgpr v -> M=v (+8 hi)
      if (gr < nrows) out[(size_t)gr * 64 + col] = acc[ct][v] + badd;
    }
  }
}

// ---------------- attention-logit reduction: alS/alD[n,h] = <xW[n,h,:], a> ----
__global__ void areduce(const float* __restrict__ xW,
                        const float* __restrict__ asrc,
                        const float* __restrict__ adst,
                        float* __restrict__ alS, float* __restrict__ alD,
                        int n, int H, int F) {
  int node = blockIdx.x * blockDim.x + threadIdx.x;
  if (node >= n) return;
  const float* row = xW + (size_t)node * 64;  // H*F == 64 always
  for (int h = 0; h < H; ++h) {
    float s = 0.f, d = 0.f;
    for (int f = 0; f < F; ++f) {
      float v = row[h * F + f];
      s += v * asrc[h * F + f];
      d += v * adst[h * F + f];
    }
    alS[node * H + h] = s;
    alD[node * H + h] = d;
  }
}

__global__ void fill_u32(unsigned* __restrict__ p, int n, unsigned v) {
  int i = blockIdx.x * blockDim.x + threadIdx.x;
  if (i < n) p[i] = v;
}

// ---------------- attention passes over edges (+N self loops on the fly) -----
__global__ void attn_max(const int* __restrict__ ei, int E, int n,
                         const float* __restrict__ alS,
                         const float* __restrict__ alD,
                         unsigned* __restrict__ mkey, int H) {
  int e = blockIdx.x * blockDim.x + threadIdx.x;
  if (e >= E + n) return;
  int src, dst;
  if (e < E) { src = ei[e]; dst = ei[E + e]; } else { src = dst = e - E; }
  for (int h = 0; h < H; ++h) {
    float v = lrelu(alS[src * H + h] + alD[dst * H + h]);
    atomicMax(&mkey[dst * H + h], fkey(v));
  }
}

__global__ void attn_den(const int* __restrict__ ei, int E, int n,
                         const float* __restrict__ alS,
                         const float* __restrict__ alD,
                         const unsigned* __restrict__ mkey,
                         float* __restrict__ den, int H) {
  int e = blockIdx.x * blockDim.x + threadIdx.x;
  if (e >= E + n) return;
  int src, dst;
  if (e < E) { src = ei[e]; dst = ei[E + e]; } else { src = dst = e - E; }
  for (int h = 0; h < H; ++h) {
    float v = lrelu(alS[src * H + h] + alD[dst * H + h]);
    float w = expf(v - funkey(mkey[dst * H + h]));
    atomicAdd(&den[dst * H + h], w);
  }
}

__global__ void attn_acc(const int* __restrict__ ei, int E, int n,
                         const float* __restrict__ alS,
                         const float* __restrict__ alD,
                         const unsigned* __restrict__ mkey,
                         const float* __restrict__ den,
                         const float* __restrict__ xW, float* __restrict__ out,
                         int H, int F) {
  int e = blockIdx.x * blockDim.x + threadIdx.x;
  if (e >= E + n) return;
  int src, dst;
  if (e < E) { src = ei[e]; dst = ei[E + e]; } else { src = dst = e - E; }
  for (int h = 0; h < H; ++h) {
    float v = lrelu(alS[src * H + h] + alD[dst * H + h]);
    float alpha = expf(v - funkey(mkey[dst * H + h])) /
                  (den[dst * H + h] + 1e-16f);
    const float* srow = xW + ((size_t)src * H + h) * F;
    float* drow = out + ((size_t)dst * H + h) * F;
    for (int f = 0; f < F; ++f) atomicAdd(&drow[f], srow[f] * alpha);
  }
}

// ---------------- bias + ELU, in place over [N,64] ----------------------------
__global__ void bias_elu(float* __restrict__ buf, const float* __restrict__ bias,
                         int total) {
  int i = blockIdx.x * blockDim.x + threadIdx.x;
  if (i >= total) return;
  float v = buf[i] + bias[i & 63];
  buf[i] = v > 0.f ? v : expm1f(v);
}

// ---------------- edge MLP: [h[src],h[dst],eattr] @ W1 -> relu -> @ W2 -> lsm -
// block = 256 threads (8 waves) handles 32 edges.
// GEMM: 32x144 @ 144x128 with 16x16x4 WMMA; wave w -> row tile (w>>2),
// col tiles 2*(w&3), 2*(w&3)+1. A gathered into LDS (stride 148: 16B-aligned
// rows, conflict-free (row*20+k) mod 64), W1 through WGP$ (72KB, L2-resident).
__global__ __launch_bounds__(256) void edge_mlp(
    const float* __restrict__ h3, const int* __restrict__ ei,
    const float* __restrict__ eattr, const float* __restrict__ W1g,
    const float* __restrict__ b1g, const float* __restrict__ W2g,
    const float* __restrict__ b2g, float* __restrict__ out, int E) {
  __shared__ float Alds[32 * 148];  // 18944 B
  __shared__ float Ylds[32 * 132];  // 16896 B
  __shared__ float W2lds[256];
  __shared__ float b1lds[128];
  const int t = threadIdx.x;
  const int e0 = blockIdx.x * 32;

  W2lds[t & 255] = W2g[t & 255];
  if (t < 128) b1lds[t] = b1g[t];

#if USE_ASYNC
  // one async b128 per 16B chunk: 36 chunks per edge row (64|64|16 split, all
  // chunk boundaries fall inside a single contiguous source region)
  for (int i = t; i < 32 * 36; i += 256) {
    int r = i / 36, c = (i - r * 36) * 4;
    int e = e0 + r;
    float* l = &Alds[r * 148 + c];
    if (e < E) {
      const float* g;
      if (c < 64)       g = &h3[(size_t)ei[e] * 64 + c];
      else if (c < 128) g = &h3[(size_t)ei[E + e] * 64 + (c - 64)];
      else              g = &eattr[(size_t)e * 16 + (c - 128)];
      async_ld16(g, l);
    } else {
      l[0] = 0.f; l[1] = 0.f; l[2] = 0.f; l[3] = 0.f;
    }
  }
  async_wait0();
#else
  for (int i = t; i < 32 * 144; i += 256) {
    int r = i / 144, c = i - r * 144;
    int e = e0 + r;
    float v = 0.f;
    if (e < E) {
      if (c < 64)        v = h3[(size_t)ei[e] * 64 + c];
      else if (c < 128)  v = h3[(size_t)ei[E + e] * 64 + (c - 64)];
      else               v = eattr[(size_t)e * 16 + (c - 128)];
    }
    Alds[r * 148 + c] = v;
  }
#endif
  __syncthreads();

  const int wave = t >> 5, lane = t & 31;
  const int half = lane >> 4, l16 = lane & 15;
  const int rt = wave >> 2;
  const int ct0 = (wave & 3) * 2;
  const int arow = rt * 16 + l16;

  v8f acc0 = {0.f, 0.f, 0.f, 0.f, 0.f, 0.f, 0.f, 0.f};
  v8f acc1 = acc0;
  for (int k0 = 0; k0 < 144; k0 += 4) {
    const int ka = k0 + half * 2;
    v2f a = {Alds[arow * 148 + ka], Alds[arow * 148 + ka + 1]};
    v2f b0 = {W1g[ka * 128 + ct0 * 16 + l16],
              W1g[(ka + 1) * 128 + ct0 * 16 + l16]};
    acc0 = wmma4(a, b0, acc0);
    v2f b1 = {W1g[ka * 128 + (ct0 + 1) * 16 + l16],
              W1g[(ka + 1) * 128 + (ct0 + 1) * 16 + l16]};
    acc1 = wmma4(a, b1, acc1);
  }

  {  // bias + relu -> Ylds
    const int c0 = ct0 * 16 + l16, c1 = (ct0 + 1) * 16 + l16;
#pragma unroll
    for (int v = 0; v < 8; ++v) {
      int row = rt * 16 + half * 8 + v;
      float y0 = acc0[v] + b1lds[c0];
      float y1 = acc1[v] + b1lds[c1];
      Ylds[row * 132 + c0] = y0 > 0.f ? y0 : 0.f;
      Ylds[row * 132 + c1] = y1 > 0.f ? y1 : 0.f;
    }
  }
  __syncthreads();

  if (t < 32) {  // 128 -> 2 + log_softmax
    int e = e0 + t;
    if (e < E) {
      float z0 = b2g[0], z1 = b2g[1];
      for (int c = 0; c < 128; ++c) {
        float y = Ylds[t * 132 + c];
        z0 += y * W2lds[c * 2 + 0];
        z1 += y * W2lds[c * 2 + 1];
      }
      float mx = fmaxf(z0, z1);
      float lse = mx + logf(expf(z0 - mx) + expf(z1 - mx));
      out[(size_t)e * 2 + 0] = z0 - lse;
      out[(size_t)e * 2 + 1] = z1 - lse;
    }
  }
}

// ------------------------------- launcher ------------------------------------
extern "C" void kernel_launch(void* const* d_in, const int* in_sizes, int n_in,
                              void* d_out, int out_size, void* d_ws,
                              size_t ws_size, hipStream_t stream) {
  const float* x     = (const float*)d_in[0];
  const int*   ei    = (const int*)d_in[1];
  const float* eattr = (const float*)d_in[2];
  const float* W1    = (const float*)d_in[3];
  const float* as1   = (const float*)d_in[4];
  const float* ad1   = (const float*)d_in[5];
  const float* b1    = (const float*)d_in[6];
  const float* W2    = (const float*)d_in[7];
  const float* as2   = (const float*)d_in[8];
  const float* ad2   = (const float*)d_in[9];
  const float* b2    = (const float*)d_in[10];
  const float* linW  = (const float*)d_in[11];
  const float* linb  = (const float*)d_in[12];
  const float* mW1   = (const float*)d_in[13];
  const float* mb1   = (const float*)d_in[14];
  const float* mW2   = (const float*)d_in[15];
  const float* mb2   = (const float*)d_in[16];
  float* out = (float*)d_out;

  const int N = in_sizes[0] / 128;
  const int E = in_sizes[1] / 2;
  const int Ep = E + N;

  float* ws   = (float*)d_ws;
  float* bufA = ws;                         // [N,64] xW scratch
  float* bufB = bufA + (size_t)N * 64;      // [N,64] out1/h1, later h3
  float* bufC = bufB + (size_t)N * 64;      // [N,64] out2/h2
  float* alS  = bufC + (size_t)N * 64;      // [N,8]
  float* alD  = alS + (size_t)N * 8;        // [N,8]
  float* den  = alD + (size_t)N * 8;        // [N,8]
  unsigned* mkey = (unsigned*)(den + (size_t)N * 8);  // [N,8]

  const int gB = (N + 63) / 64;              // gemm blocks
  const int nB = (N + 255) / 256;            // node-wise blocks
  const int eB = (Ep + 255) / 256;           // edge-wise blocks (w/ self loops)
  const int fB16 = (N * 16 + 255) / 256;     // den+mkey fill
  const int fB64 = (N * 64 + 255) / 256;     // [N,64] fill / elementwise

  // ---- GAT layer 1 ----
  gemm_k_n64<<<gB, 128, 0, stream>>>(x, W1, nullptr, bufA, N, 128);
  areduce<<<nB, 256, 0, stream>>>(bufA, as1, ad1, alS, alD, N, 8, 8);
  fill_u32<<<fB16, 256, 0, stream>>>((unsigned*)den, N * 16, 0u);
  fill_u32<<<fB64, 256, 0, stream>>>((unsigned*)bufB, N * 64, 0u);
  attn_max<<<eB, 256, 0, stream>>>(ei, E, N, alS, alD, mkey, 8);
  attn_den<<<eB, 256, 0, stream>>>(ei, E, N, alS, alD, mkey, den, 8);
  attn_acc<<<eB, 256, 0, stream>>>(ei, E, N, alS, alD, mkey, den, bufA, bufB, 8, 8);
  bias_elu<<<fB64, 256, 0, stream>>>(bufB, b1, N * 64);

  // ---- GAT layer 2 (heads=1, F=64) ----
  gemm_k_n64<<<gB, 128, 0, stream>>>(bufB, W2, nullptr, bufA, N, 64);
  areduce<<<nB, 256, 0, stream>>>(bufA, as2, ad2, alS, alD, N, 1, 64);
  fill_u32<<<fB16, 256, 0, stream>>>((unsigned*)den, N * 16, 0u);
  fill_u32<<<fB64, 256, 0, stream>>>((unsigned*)bufC, N * 64, 0u);
  attn_max<<<eB, 256, 0, stream>>>(ei, E, N, alS, alD, mkey, 1);
  attn_den<<<eB, 256, 0, stream>>>(ei, E, N, alS, alD, mkey, den, 1);
  attn_acc<<<eB, 256, 0, stream>>>(ei, E, N, alS, alD, mkey, den, bufA, bufC, 1, 64);
  bias_elu<<<fB64, 256, 0, stream>>>(bufC, b2, N * 64);

  // ---- node linear: h3 = h2 @ lin_W + lin_b ----
  gemm_k_n64<<<gB, 128, 0, stream>>>(bufC, linW, linb, bufB, N, 64);

  // ---- edge MLP + log_softmax ----
  edge_mlp<<<(E + 31) / 32, 256, 0, stream>>>(bufB, ei, eattr, mW1, mb1, mW2,
                                              mb2, out, E);
}